// Network_32358283608359
// MI455X (gfx1250) — compile-verified
//
#include <hip/hip_runtime.h>
#include <hip/hip_bf16.h>
#include <math.h>

// ---------------------------------------------------------------------------
// Plastic-RNN step, H=4096, bandwidth-bound (~256MB HBM traffic/call,
// ~11-14us at 23.3 TB/s).
// Pass 1: stream w/alpha/hebb once (w,alpha non-temporal; hebb L2-resident,
//         64MB << 192MB L2), GEMV hidden @ (w + alpha*hebb) via
//         V_WMMA_F32_16X16X4_F32 (full f32 precision, compute is free here).
// Pass 2: hactiv = tanh(x@Wi + bi + r)
// Pass 3: hebb_new = (1-eta)*hebb + eta*outer(hidden,hactiv)
//         (hebb re-read from L2; NT store of result — write-once, never read)
// Pass 4: softmax head + value head
// ---------------------------------------------------------------------------

#define H 4096
#define HH (4096ull * 4096ull)
#define KSPLIT 32                 // K-splits in pass 1 (waves = 256*KSPLIT)

typedef __attribute__((ext_vector_type(2))) float v2f;
typedef __attribute__((ext_vector_type(8))) float v8f;

// ---------------------------------------------------------------------------
__global__ void k0_zero_r(float* __restrict__ r) {
    int j = blockIdx.x * blockDim.x + threadIdx.x;
    if (j < H) r[j] = 0.0f;
}

// ---------------------------------------------------------------------------
// Pass 1: r[col] += sum_k hidden[k] * (w[k][col] + alpha[k][col]*hebb[k][col])
// 8192 waves: wid%256 -> 16-column group, wid/256 -> 128-row K-split.
// Each wave: 32 iterations of V_WMMA_F32_16X16X4_F32 (K=4 per step).
//
// A (16x4 f32) layout: VGPR0 = K0 (lanes0-15) / K2 (lanes16-31); VGPR1 = K1/K3.
//   Only M-row 0 nonzero -> lane0 {h[k],h[k+1]}, lane16 {h[k+2],h[k+3]}.
// B (4x16 f32) layout: VGPR0 = row K0 (lanes0-15,N=lane) / row K2 (lanes16-31);
//   VGPR1 = rows K1 / K3.
// D (16x16 f32): D[0][n] lives in c[0], lanes 0-15.
// ---------------------------------------------------------------------------
__global__ void k1_gemv_wmma(const float* __restrict__ hidden,
                             const float* __restrict__ hebb,
                             const float* __restrict__ w,
                             const float* __restrict__ alpha,
                             float* __restrict__ r) {
    const int lane    = threadIdx.x & 31;
    const int wid     = (blockIdx.x * blockDim.x + threadIdx.x) >> 5; // 0..256*KSPLIT-1
    const int colgrp  = wid & 255;            // 256 groups of 16 columns
    const int kblk    = wid >> 8;             // KSPLIT splits of H/KSPLIT rows
    const int halfsel = lane >> 4;            // 0: lanes 0-15, 1: lanes 16-31
    const int nlane   = lane & 15;
    const int col     = colgrp * 16 + nlane;
    const int k0      = kblk * (H / KSPLIT);
    const int kend    = k0 + (H / KSPLIT);

    v8f c = {};  // accumulator; rows 1..15 stay zero (A rows 1..15 are zero)
    size_t off = (size_t)(k0 + 2 * halfsel) * H + (size_t)col;

    for (int k = k0; k < kend; k += 4) {
        const int hidx = k + 2 * halfsel;
        // A-matrix: row 0 only
        v2f A;
        A.x = (nlane == 0) ? hidden[hidx]     : 0.0f;
        A.y = (nlane == 0) ? hidden[hidx + 1] : 0.0f;
        // B-matrix: M = w + alpha*hebb, rows {k+2h, k+1+2h} at this lane's col.
        // w/alpha are single-use -> non-temporal; hebb cached (L2) for pass 3.
        const float w0 = __builtin_nontemporal_load(&w[off]);
        const float a0 = __builtin_nontemporal_load(&alpha[off]);
        const float w1 = __builtin_nontemporal_load(&w[off + H]);
        const float a1 = __builtin_nontemporal_load(&alpha[off + H]);
        v2f B;
        B.x = fmaf(a0, hebb[off], w0);
        B.y = fmaf(a1, hebb[off + H], w1);
        // D = A x B + C   (f32, 16x16x4)
        c = __builtin_amdgcn_wmma_f32_16x16x4_f32(false, A, false, B,
                                                  (short)0, c, false, false);
        off += (size_t)4 * H;
    }

    if (lane < 16) {
        atomicAdd(&r[col], c[0]);   // D[0][lane] for this column group
    }
}

// ---------------------------------------------------------------------------
// Pass 2: hactiv[j] = tanh( sum_t x[t]*Wi[t][j] + bi[j] + r[j] )
// ---------------------------------------------------------------------------
__global__ void k2_hactiv(const float* __restrict__ x,
                          const float* __restrict__ Wi,
                          const float* __restrict__ bi,
                          const float* __restrict__ r,
                          float* __restrict__ hact_out,
                          float* __restrict__ hact_ws) {
    int j = blockIdx.x * blockDim.x + threadIdx.x;
    if (j >= H) return;
    float acc = bi[j] + r[j];
#pragma unroll
    for (int t = 0; t < 17; ++t) {
        acc = fmaf(x[t], Wi[t * H + j], acc);
    }
    float h = tanhf(acc);
    hact_out[j] = h;   // d_out region (offset 5, only 4B-aligned)
    hact_ws[j]  = h;   // aligned scratch copy for passes 3/4
}

// ---------------------------------------------------------------------------
// Pass 3: hebb_new[i][j] = (1-eta)*hebb[i][j] + eta*hidden[i]*hactiv[j]
// hebb re-read hits L2 (RT lines from pass 1); this is its last use -> NT.
// Result is write-once/never-read -> NT store (don't rinse L2).
// Output region is only 4-byte aligned -> coalesced scalar f32 grid-stride.
// ---------------------------------------------------------------------------
__global__ void k3_hebb(const float* __restrict__ hebb,
                        const float* __restrict__ hidden,
                        const float* __restrict__ hact,
                        const float* __restrict__ eta_p,
                        float* __restrict__ out) {
    const float eta = eta_p[0];
    const float om  = 1.0f - eta;
    const size_t stride = (size_t)gridDim.x * blockDim.x;
    for (size_t e = (size_t)blockIdx.x * blockDim.x + threadIdx.x; e < HH; e += stride) {
        const int i = (int)(e >> 12);
        const int j = (int)(e & (H - 1));
        const float hb = __builtin_nontemporal_load(&hebb[e]);
        __builtin_nontemporal_store(fmaf(om, hb, eta * hidden[i] * hact[j]), &out[e]);
    }
}

// ---------------------------------------------------------------------------
// Pass 4: activout = softmax(hactiv @ Wo + bo); valueout = hactiv @ Wv + bv
// Single block, LDS tree reduction over 5 partials.
// ---------------------------------------------------------------------------
__global__ void k4_heads(const float* __restrict__ hact,
                         const float* __restrict__ Wo,
                         const float* __restrict__ bo,
                         const float* __restrict__ Wv,
                         const float* __restrict__ bv,
                         float* __restrict__ out) {
    __shared__ float red[5 * 256];
    const int t = threadIdx.x;
    float p0 = 0.f, p1 = 0.f, p2 = 0.f, p3 = 0.f, pv = 0.f;
    for (int j = t; j < H; j += 256) {
        const float h = hact[j];
        p0 = fmaf(h, Wo[j * 4 + 0], p0);
        p1 = fmaf(h, Wo[j * 4 + 1], p1);
        p2 = fmaf(h, Wo[j * 4 + 2], p2);
        p3 = fmaf(h, Wo[j * 4 + 3], p3);
        pv = fmaf(h, Wv[j], pv);
    }
    red[0 * 256 + t] = p0;
    red[1 * 256 + t] = p1;
    red[2 * 256 + t] = p2;
    red[3 * 256 + t] = p3;
    red[4 * 256 + t] = pv;
    __syncthreads();
    for (int s = 128; s > 0; s >>= 1) {
        if (t < s) {
#pragma unroll
            for (int q = 0; q < 5; ++q) red[q * 256 + t] += red[q * 256 + t + s];
        }
        __syncthreads();
    }
    if (t == 0) {
        const float l0 = red[0] + bo[0];
        const float l1 = red[256] + bo[1];
        const float l2 = red[512] + bo[2];
        const float l3 = red[768] + bo[3];
        const float m  = fmaxf(fmaxf(l0, l1), fmaxf(l2, l3));
        const float e0 = expf(l0 - m), e1 = expf(l1 - m);
        const float e2 = expf(l2 - m), e3 = expf(l3 - m);
        const float inv = 1.0f / (e0 + e1 + e2 + e3);
        out[0] = e0 * inv;
        out[1] = e1 * inv;
        out[2] = e2 * inv;
        out[3] = e3 * inv;
        out[4] = red[4 * 256] + bv[0];
    }
}

// ---------------------------------------------------------------------------
extern "C" void kernel_launch(void* const* d_in, const int* in_sizes, int n_in,
                              void* d_out, int out_size, void* d_ws, size_t ws_size,
                              hipStream_t stream) {
    // setup_inputs() order:
    const float* x      = (const float*)d_in[0];   // [1,17]
    const float* hidden = (const float*)d_in[1];   // [1,4096]
    const float* hebb   = (const float*)d_in[2];   // [4096,4096]
    const float* Wi     = (const float*)d_in[3];   // [17,4096]
    const float* bi     = (const float*)d_in[4];   // [4096]
    const float* w      = (const float*)d_in[5];   // [4096,4096]
    const float* alpha  = (const float*)d_in[6];   // [4096,4096]
    const float* eta    = (const float*)d_in[7];   // [1]
    const float* Wo     = (const float*)d_in[8];   // [4096,4]
    const float* bo     = (const float*)d_in[9];   // [4]
    const float* Wv     = (const float*)d_in[10];  // [4096,1]
    const float* bv     = (const float*)d_in[11];  // [1]

    float* out = (float*)d_out;
    float* out_activ = out + 0;      // [4]
    // out+4 = valueout [1] (written by k4)
    float* out_hact  = out + 5;      // [4096]
    float* out_hebb  = out + 4101;   // [4096*4096]

    float* r       = (float*)d_ws;            // [4096] partial GEMV sums
    float* hact_ws = (float*)d_ws + H;        // [4096] aligned hactiv copy

    // Pass 0: zero partials (deterministic each call)
    k0_zero_r<<<(H + 255) / 256, 256, 0, stream>>>(r);

    // Pass 1: WMMA GEMV over M = w + alpha*hebb.
    // 256*KSPLIT waves, 8 waves/block (256 threads).
    k1_gemv_wmma<<<(256 * KSPLIT) / 8, 256, 0, stream>>>(hidden, hebb, w, alpha, r);

    // Pass 2: tanh activation
    k2_hactiv<<<(H + 255) / 256, 256, 0, stream>>>(x, Wi, bi, r, out_hact, hact_ws);

    // Pass 3: hebb trace update (16.7M elems, grid-stride)
    k3_hebb<<<8192, 256, 0, stream>>>(hebb, hidden, hact_ws, eta, out_hebb);

    // Pass 4: softmax + value heads
    k4_heads<<<1, 256, 0, stream>>>(hact_ws, Wo, bo, Wv, bv, out_activ);
}